// RoIAlign3D_33423435498477
// MI455X (gfx1250) — compile-verified
//
#include <hip/hip_runtime.h>
#include <hip/hip_bf16.h>

typedef float v2f __attribute__((ext_vector_type(2)));
typedef float v8f __attribute__((ext_vector_type(8)));

namespace {
constexpr int Cc = 128, Td = 16, Hd = 56, Wd = 56;
constexpr int HW = Hd * Wd, THW = Td * HW;
constexpr int oT = 4, oH = 8, oW = 8;
constexpr float Tsc = 0.25f, Ssc = 0.0625f;
}

__device__ __forceinline__ void axis_interp(float coord, int size,
                                            int& lo, int& hi,
                                            float& wl, float& wh, bool& valid) {
  valid = (coord >= -1.0f) && (coord <= (float)size);
  float c = fminf(fmaxf(coord, 0.0f), (float)(size - 1));
  int l = (int)floorf(c);
  if (l > size - 1) l = size - 1;
  int h = l + 1;
  if (h > size - 1) h = size - 1;
  float f = c - (float)l;
  lo = l; hi = h; wl = 1.0f - f; wh = f;
}

// One block per (roi, oy, ox); 8 waves = 8 channel tiles of 16.
// Each wave computes a 16(channel) x 4(ot) tile via 16 chained
// v_wmma_f32_16x16x4_f32 ops: K = 16 temporal-level slots, C-accumulation
// over the 4 in-bin spatial samples. Temporal weights (+validity, +1/8
// average) live in the shared B matrix; bilinear xy interpolation (with
// spatial validity) is folded into the gathered A matrix.
// All feature gathers use a uniform base pointer + 32-bit per-lane index
// (tensor < 2^31 bytes) so the backend can emit saddr-form global loads.
__global__ __launch_bounds__(256)
void roialign3d_wmma_kern(const float* __restrict__ feat,
                          const float* __restrict__ rois,
                          float* __restrict__ out) {
  const int blk  = blockIdx.x;
  const int r    = blk >> 6;
  const int bin  = blk & 63;
  const int oy   = bin >> 3;
  const int ox   = bin & 7;
  const int wave = threadIdx.x >> 5;
  const int lane = threadIdx.x & 31;
  const int half = lane >> 4;   // K-half for A/B, row-half for D
  const int m    = lane & 15;   // A row (channel-in-tile) / B,D column

  const float* rp = rois + r * 7;
  const int   b  = (int)rp[0];
  const float ts = rp[1] * Tsc, te = rp[4] * Tsc;
  const float xs = rp[2] * Ssc, xe = rp[5] * Ssc;
  const float ys = rp[3] * Ssc, ye = rp[6] * Ssc;
  const float bt = fmaxf(te - ts, 1.0f) * (1.0f / oT);
  const float by = fmaxf(ye - ys, 1.0f) * (1.0f / oH);
  const float bx = fmaxf(xe - xs, 1.0f) * (1.0f / oW);

  // Temporal axis: 8 sample coords -> low/high level + weight (x vt x 1/8).
  int Tlo[8], Thi[8];
  float Wlo[8], Whi[8];
#pragma unroll
  for (int j = 0; j < 8; ++j) {
    float tc = ts + ((float)j * 0.5f + 0.25f) * bt;
    int lo, hi; float wl, wh; bool v;
    axis_interp(tc, Td, lo, hi, wl, wh, v);
    Tlo[j] = lo; Thi[j] = hi;
    float s = v ? 0.125f : 0.0f;  // fold 1/(S^3) average + t-validity
    Wlo[j] = wl * s; Whi[j] = wh * s;
  }

  // B matrix (K=16 x N=16, cols 0..3 = ot) in 4 chunks of K=4, plus the
  // t-level each lane's two K slots need. Slot k: k<8 -> low of t-sample k,
  // k>=8 -> high of t-sample k-8; slot k maps to column (k&7)>>1.
  // Layout assumption (symmetric with documented 16x4 A layout):
  // lanes 0-15 carry K-local {0,1}, lanes 16-31 carry K-local {2,3}.
  const int c     = wave * 16 + m;
  const int off_c = (b * Cc + c) * THW;  // < 2^25 elements, fits in int

  v2f Bk[4];
  int TO0[4], TO1[4];  // per-chunk temporal plane offsets (elements)
#pragma unroll
  for (int kk = 0; kk < 4; ++kk) {
    const bool hiK = (kk >= 2);
    const int  jb  = (kk & 1) * 4;  // compile-time per unroll step
    float ws0 = half ? (hiK ? Whi[jb + 2] : Wlo[jb + 2])
                     : (hiK ? Whi[jb + 0] : Wlo[jb + 0]);
    float ws1 = half ? (hiK ? Whi[jb + 3] : Wlo[jb + 3])
                     : (hiK ? Whi[jb + 1] : Wlo[jb + 1]);
    int t0 = half ? (hiK ? Thi[jb + 2] : Tlo[jb + 2])
                  : (hiK ? Thi[jb + 0] : Tlo[jb + 0]);
    int t1 = half ? (hiK ? Thi[jb + 3] : Tlo[jb + 3])
                  : (hiK ? Thi[jb + 1] : Tlo[jb + 1]);
    const int col = (2 * kk + half) & 3;  // both slots of this lane hit one col
    Bk[kk].x = (m == col) ? ws0 : 0.0f;
    Bk[kk].y = (m == col) ? ws1 : 0.0f;
    TO0[kk] = off_c + t0 * HW;
    TO1[kk] = off_c + t1 * HW;
  }

  // Spatial axes for this bin: 2 y-samples, 2 x-samples.
  int Ylo[2], Yhi[2], Xlo[2], Xhi[2];
  float Wyl[2], Wyh[2], Wxl[2], Wxh[2];
  bool Vy[2], Vx[2];
#pragma unroll
  for (int j = 0; j < 2; ++j) {
    float yc = ys + ((float)(oy * 2 + j) * 0.5f + 0.25f) * by;
    axis_interp(yc, Hd, Ylo[j], Yhi[j], Wyl[j], Wyh[j], Vy[j]);
    float xc = xs + ((float)(ox * 2 + j) * 0.5f + 0.25f) * bx;
    axis_interp(xc, Wd, Xlo[j], Xhi[j], Wxl[j], Wxh[j], Vx[j]);
  }

  v8f acc = {0.0f, 0.0f, 0.0f, 0.0f, 0.0f, 0.0f, 0.0f, 0.0f};
#pragma unroll
  for (int jy = 0; jy < 2; ++jy) {
#pragma unroll
    for (int jx = 0; jx < 2; ++jx) {
      const bool  sv  = Vy[jy] && Vx[jx];
      const float wll = sv ? Wyl[jy] * Wxl[jx] : 0.0f;
      const float wlh = sv ? Wyl[jy] * Wxh[jx] : 0.0f;
      const float whl = sv ? Wyh[jy] * Wxl[jx] : 0.0f;
      const float whh = sv ? Wyh[jy] * Wxh[jx] : 0.0f;
      const int oll = Ylo[jy] * Wd + Xlo[jx];
      const int olh = Ylo[jy] * Wd + Xhi[jx];
      const int ohl = Yhi[jy] * Wd + Xlo[jx];
      const int ohh = Yhi[jy] * Wd + Xhi[jx];
#pragma unroll
      for (int kk = 0; kk < 4; ++kk) {
        const int t0 = TO0[kk];
        const int t1 = TO1[kk];
        v2f A;
        A.x = wll * feat[t0 + oll] + wlh * feat[t0 + olh] +
              whl * feat[t0 + ohl] + whh * feat[t0 + ohh];
        A.y = wll * feat[t1 + oll] + wlh * feat[t1 + olh] +
              whl * feat[t1 + ohl] + whh * feat[t1 + ohh];
        acc = __builtin_amdgcn_wmma_f32_16x16x4_f32(
            false, A, false, Bk[kk], (short)0, acc, false, false);
      }
    }
  }

  // D layout: lane holds column N=m; VGPR v holds row M = v + 8*half.
  // Columns 0..3 carry ot; store 16 channels x 4 ot per wave.
  if (m < 4) {
    const int ot = m;
    const int sp = oy * oW + ox;
#pragma unroll
    for (int v = 0; v < 8; ++v) {
      const int cc = wave * 16 + 8 * half + v;
      out[((r * Cc + cc) * oT + ot) * (oH * oW) + sp] = acc[v];
    }
  }
}

extern "C" void kernel_launch(void* const* d_in, const int* in_sizes, int n_in,
                              void* d_out, int out_size, void* d_ws,
                              size_t ws_size, hipStream_t stream) {
  const float* feat = (const float*)d_in[0];
  const float* rois = (const float*)d_in[1];
  float* out = (float*)d_out;
  const int R = in_sizes[1] / 7;  // rois is (R, 7)
  dim3 grid(R * (oH * oW));       // one block per (roi, oy, ox)
  roialign3d_wmma_kern<<<grid, 256, 0, stream>>>(feat, rois, out);
}